// DynamicsBasic_56495999812284
// MI455X (gfx1250) — compile-verified
//
#include <hip/hip_runtime.h>
#include <hip/hip_bf16.h>

// y = (W @ x_i) . (x_f - x_i)  ==  sum_j x_i[j] * (W^T d)[j],  d = x_f - x_i
// N = 8192, W fp32 row-major, 256 MB streamed exactly once (HBM-bound, ~11us
// floor at 23.3 TB/s). Compute engine: V_WMMA_F32_16X16X4_F32 (exact fp32).
// d is precomputed once into d_ws; main loop = 1 b64 (d) + 2 b32 (W) loads
// per WMMA, dual accumulators to break the WMMA C-chain dependency.

typedef __attribute__((ext_vector_type(2))) float v2f;
typedef __attribute__((ext_vector_type(8))) float v8f;

#define NDIM 8192
#define COLS_PER_TILE 16
#define ROW_SEGS 32
#define ROWS_PER_SEG (NDIM / ROW_SEGS)          // 256 rows per wave
#define WAVES_PER_BLOCK 8
#define THREADS (WAVES_PER_BLOCK * 32)          // 256 threads = 8 wave32
#define COL_TILES (NDIM / COLS_PER_TILE)        // 512
#define TOTAL_WAVES (COL_TILES * ROW_SEGS)      // 16384
#define GRID (TOTAL_WAVES / WAVES_PER_BLOCK)    // 2048 blocks

// d_ws[0..N-1] = x_f - x_i ; also zero the scalar output (stream-ordered).
__global__ __launch_bounds__(256)
void dyn_prep(const float* __restrict__ x_i, const float* __restrict__ x_f,
              float* __restrict__ d, float* __restrict__ out) {
    int i = blockIdx.x * 256 + threadIdx.x;
    d[i] = x_f[i] - x_i[i];
    if (i == 0) out[0] = 0.0f;
}

__global__ __launch_bounds__(THREADS)
void dyn_bilinear_wmma(const float* __restrict__ x_i,
                       const float* __restrict__ dvec,
                       const float* __restrict__ W,
                       float* __restrict__ out) {
    const int lane   = threadIdx.x & 31;
    const int waveId = blockIdx.x * WAVES_PER_BLOCK + (threadIdx.x >> 5);
    const int coltile = waveId % COL_TILES;      // which 16 columns of W
    const int rowseg  = waveId / COL_TILES;      // which 256-row segment
    const int colbase = coltile * COLS_PER_TILE;
    const int row0    = rowseg * ROWS_PER_SEG;

    const int hi   = lane >> 4;     // 0: lanes 0-15 (low K half), 1: lanes 16-31
    const int nIdx = lane & 15;     // column within tile (B/D layout: N = lane%16)

    // B layout 4x16 f32 (lanes 16-31 hold upper K half, one K-row per VGPR):
    // VGPR0 = rows {k0, k0+2}, VGPR1 = rows {k0+1, k0+3}.
    const float* wPtr0 = W + (size_t)(row0 + 2 * hi) * NDIM + colbase + nIdx;
    const float* wPtr1 = wPtr0 + (size_t)4 * NDIM;
    // A layout 16x4 f32: VGPR0 = K{0,2}, VGPR1 = K{1,3} split across lane
    // halves, broadcast over M: a = {d[k0+2*hi], d[k0+2*hi+1]}.
    const float* dP0 = dvec + row0 + 2 * hi;
    const float* dP1 = dP0 + 4;

    v8f acc0 = {};
    v8f acc1 = {};
    #pragma unroll 2
    for (int k = 0; k < ROWS_PER_SEG; k += 8) {
        v2f a0 = *(const v2f*)dP0;
        v2f a1 = *(const v2f*)dP1;
        v2f b0, b1;
        b0.x = wPtr0[0];            // W[k0 + 2*hi    ][colbase + nIdx]
        b0.y = wPtr0[NDIM];         // W[k0 + 2*hi + 1][colbase + nIdx]
        b1.x = wPtr1[0];
        b1.y = wPtr1[NDIM];
        // D[m][n] += sum_k d[k0+k] * W[k0+k][colbase+n]  (exact fp32 FMA tree)
        acc0 = __builtin_amdgcn_wmma_f32_16x16x4_f32(
            false, a0, false, b0, (short)0, acc0, false, false);
        acc1 = __builtin_amdgcn_wmma_f32_16x16x4_f32(
            false, a1, false, b1, (short)0, acc1, false, false);
        wPtr0 += (size_t)8 * NDIM;
        wPtr1 += (size_t)8 * NDIM;
        dP0 += 8;
        dP1 += 8;
    }

    // acc[0] at lane l holds u_part[colbase + l%16], replicated across M and
    // across lane halves -> sum all 32 lanes, scale by exactly 0.5.
    float contrib = (acc0[0] + acc1[0]) * x_i[colbase + nIdx];
    #pragma unroll
    for (int off = 16; off > 0; off >>= 1)
        contrib += __shfl_xor(contrib, off, 32);
    if (lane == 0) atomicAdd(out, 0.5f * contrib);
}

extern "C" void kernel_launch(void* const* d_in, const int* in_sizes, int n_in,
                              void* d_out, int out_size, void* d_ws, size_t ws_size,
                              hipStream_t stream) {
    const float* x_i = (const float*)d_in[0];
    const float* x_f = (const float*)d_in[1];
    const float* W   = (const float*)d_in[2];
    // d_in[3] = t   : unused (d interpolate/dt == x_f - x_i, independent of t)
    // d_in[4] = arg : unused
    float* out  = (float*)d_out;
    float* dvec = (float*)d_ws;   // 8192 floats = 32 KB scratch

    dyn_prep<<<NDIM / 256, 256, 0, stream>>>(x_i, x_f, dvec, out);
    dyn_bilinear_wmma<<<GRID, THREADS, 0, stream>>>(x_i, dvec, W, out);
}